// ROIAlign_47802986005225
// MI455X (gfx1250) — compile-verified
//
#include <hip/hip_runtime.h>

// ---------------------------------------------------------------------------
// ROI Align (FPN, P=7, SR=2) for MI455X / gfx1250.
// Gather-bound op; pyramid (~170MB) is L2-resident (192MB L2).
// This version stages each channel's ROI bounding rectangle into LDS with
// CDNA5 async global->LDS DMA (global_load_async_to_lds_b128), double
// buffered one channel ahead (s_wait_asynccnt + split barrier), then does the
// 16-term bilinear/pool reduction out of LDS. NT stores keep L2 for features.
// ---------------------------------------------------------------------------

#define POOL 7
#define NCH  256
#define LSTR 40      // LDS tile row stride in floats (mult of 4)
#define TROWS 34     // max rows in ROI rectangle (spanY <= 33)

typedef int v4ig __attribute__((vector_size(16)));          // matches builtin V4i
typedef unsigned int u32x4 __attribute__((vector_size(16)));
typedef int i32x4 __attribute__((vector_size(16)));
typedef int i32x8 __attribute__((vector_size(32)));

static __device__ __forceinline__ void async_cp16(const float* g, float* l)
{
#if __has_builtin(__builtin_amdgcn_global_load_async_to_lds_b128)
    __builtin_amdgcn_global_load_async_to_lds_b128(
        (__attribute__((address_space(1))) v4ig*)g,
        (__attribute__((address_space(3))) v4ig*)l,
        0, 0);
#else
    *(v4ig*)l = *(const v4ig*)g;   // sync fallback
#endif
}

static __device__ __forceinline__ void wait_async0()
{
#if __has_builtin(__builtin_amdgcn_s_wait_asynccnt)
    __builtin_amdgcn_s_wait_asynccnt(0);
#else
    asm volatile("s_wait_asynccnt 0" ::: "memory");
#endif
}

__global__ void __launch_bounds__(64)
roi_align_fwd(const float* __restrict__ f0, const float* __restrict__ f1,
              const float* __restrict__ f2, const float* __restrict__ f3,
              const float* __restrict__ boxes,
              const int* __restrict__ image_ids,
              const int* __restrict__ level_ids,
              float* __restrict__ out)
{
    __shared__ __align__(16) float tile[2][TROWS * LSTR];   // 2 x 5440 B

    const int n     = blockIdx.x;      // box id
    const int chunk = blockIdx.y;      // 4 chunks of 64 channels
    const int t     = threadIdx.x;

    const int lvl = __builtin_amdgcn_readfirstlane(level_ids[n]);
    const int img = __builtin_amdgcn_readfirstlane(image_ids[n]);

    const float* feat; int H; float inv_s;
    if (lvl == 0)      { feat = f0; H = 256; inv_s = 0.25f;    }
    else if (lvl == 1) { feat = f1; H = 128; inv_s = 0.125f;   }
    else if (lvl == 2) { feat = f2; H = 64;  inv_s = 0.0625f;  }
    else               { feat = f3; H = 32;  inv_s = 0.03125f; }
    const int W  = H;
    const int HW = H * W;
    const float* base0 = feat + (size_t)(img * NCH + chunk * 64) * (size_t)HW;

    const float x1 = boxes[4 * n + 0] * inv_s;
    const float y1 = boxes[4 * n + 1] * inv_s;
    const float x2 = boxes[4 * n + 2] * inv_s;
    const float y2 = boxes[4 * n + 3] * inv_s;
    const float bw = fmaxf(x2 - x1, 1.0f) * (1.0f / (float)POOL);
    const float bh = fmaxf(y2 - y1, 1.0f) * (1.0f / (float)POOL);

    // ---- ROI bounding rectangle (uniform across block; samples monotone) ----
    // sample positions: pos_i = (i+0.5)/2, i = 0..13  ->  first 0.25, last 6.75
    const float ysf = y1 + 0.25f * bh, ysl = y1 + 6.75f * bh;
    const float xsf = x1 + 0.25f * bw, xsl = x1 + 6.75f * bw;
    const int ry0 = (int)fminf(fmaxf(ysf, 0.0f), (float)(H - 1));
    const int ry1 = min((int)fminf(fmaxf(ysl, 0.0f), (float)(H - 1)) + 1, H - 1);
    const int rx0 = (int)fminf(fmaxf(xsf, 0.0f), (float)(W - 1));
    const int rx1 = min((int)fminf(fmaxf(xsl, 0.0f), (float)(W - 1)) + 1, W - 1);

    const int rx0a  = rx0 & ~3;                    // 16B-aligned start col
    const int kx    = (rx1 - rx0a + 4) >> 2;       // 16B chunks per row
    const int rxe   = min(rx0a + 4 * kx, W);       // exclusive end col (mult 4)
    const int spanY = ry1 - ry0 + 1;               // <= 33
    const int ntasks = spanY * kx;                 // <= 33*9 = 297

    // per-lane 2D task walk without per-iteration division
    const int q = 64 / kx, rem = 64 % kx;
    int r0 = t / kx, c0 = t - r0 * kx;

    auto issue_copy = [&](int buf, const float* plane) {
        int r = r0, c = c0;
        for (int j = t; j < ntasks; j += 64) {
            int col = rx0a + (c << 2);
            if (col > rxe - 4) col = rxe - 4;      // back-shift: stay in row
            async_cp16(plane + (ry0 + r) * W + col,
                       &tile[buf][r * LSTR + (col - rx0a)]);
            c += rem; r += q;
            if (c >= kx) { c -= kx; r += 1; }
        }
    };

    // ---- per-bin geometry: 16 (LDS offset, weight) pairs ----
    const bool active = (t < POOL * POOL);
    const int  py = active ? (t / POOL) : 0;
    const int  px = active ? (t % POOL) : 0;

    int   yl[2], yh[2], xl[2], xh[2];
    float fy[2], fx[2];
    bool  vy[2], vx[2];
#pragma unroll
    for (int d = 0; d < 2; ++d) {
        float y = y1 + ((float)py + ((float)d + 0.5f) * 0.5f) * bh;
        vy[d]   = (y > -1.0f) && (y < (float)H);
        float yc = fminf(fmaxf(y, 0.0f), (float)(H - 1));
        int   lo = (int)yc;
        yl[d] = lo; yh[d] = min(lo + 1, H - 1); fy[d] = yc - (float)lo;

        float x = x1 + ((float)px + ((float)d + 0.5f) * 0.5f) * bw;
        vx[d]   = (x > -1.0f) && (x < (float)W);
        float xc = fminf(fmaxf(x, 0.0f), (float)(W - 1));
        int   lx = (int)xc;
        xl[d] = lx; xh[d] = min(lx + 1, W - 1); fx[d] = xc - (float)lx;
    }

    int   loff[16];
    float wgt[16];
#pragma unroll
    for (int dy = 0; dy < 2; ++dy) {
#pragma unroll
        for (int dx = 0; dx < 2; ++dx) {
            const float m  = (vy[dy] && vx[dx]) ? 0.25f : 0.0f;
            const int   k  = (dy * 2 + dx) * 4;
            const int   rl = (yl[dy] - ry0) * LSTR, rh = (yh[dy] - ry0) * LSTR;
            const int   cl = xl[dx] - rx0a, ch = xh[dx] - rx0a;
            const float hy = 1.0f - fy[dy], ly = fy[dy];
            const float hx = 1.0f - fx[dx], lx_ = fx[dx];
            loff[k + 0] = rl + cl; wgt[k + 0] = m * hy * hx;
            loff[k + 1] = rl + ch; wgt[k + 1] = m * hy * lx_;
            loff[k + 2] = rh + cl; wgt[k + 2] = m * ly * hx;
            loff[k + 3] = rh + ch; wgt[k + 3] = m * ly * lx_;
        }
    }

    // ---- channel loop: double-buffered async tile, compute from LDS ----
    const size_t obase = ((size_t)n * NCH + (size_t)chunk * 64) * 49 + (size_t)t;

    issue_copy(0, base0);                           // prologue: channel 0
    for (int ci = 0; ci < 64; ++ci) {
        wait_async0();                              // my async writes landed
        __syncthreads();                            // everyone's writes landed
        if (ci + 1 < 64) {
            issue_copy((ci + 1) & 1, base0 + (size_t)(ci + 1) * HW);
            __builtin_prefetch(base0 + (size_t)(ci + 2) * HW + ry0 * W + rx0a, 0, 3);
        }
        if (active) {
            const float* tb = tile[ci & 1];
            float acc = 0.0f;
#pragma unroll
            for (int k = 0; k < 16; ++k)
                acc = fmaf(wgt[k], tb[loff[k]], acc);
            __builtin_nontemporal_store(acc, &out[obase + (size_t)ci * 49]);
        }
    }
}

// ---------------------------------------------------------------------------
// TDM probe (never launched): confirm 6-arg tensor_load_to_lds lowers, so a
// descriptor-based tile copy could replace the async loads in a later round.
// ---------------------------------------------------------------------------
__global__ void cdna5_tdm_probe(float* __restrict__ o)
{
    __shared__ __align__(16) float tl[64];
#if __has_builtin(__builtin_amdgcn_tensor_load_to_lds)
    u32x4 g0 = {}; i32x8 g1 = {}; i32x4 g2 = {}; i32x4 g3 = {}; i32x8 g4 = {};
    __builtin_amdgcn_tensor_load_to_lds(g0, g1, g2, g3, g4, 0);
#endif
#if __has_builtin(__builtin_amdgcn_s_wait_tensorcnt)
    __builtin_amdgcn_s_wait_tensorcnt(0);
#endif
    __syncthreads();
    o[threadIdx.x] = tl[threadIdx.x & 63];
}

// ---------------------------------------------------------------------------
extern "C" void kernel_launch(void* const* d_in, const int* in_sizes, int n_in,
                              void* d_out, int out_size, void* d_ws, size_t ws_size,
                              hipStream_t stream)
{
    const float* f0        = (const float*)d_in[0];
    const float* f1        = (const float*)d_in[1];
    const float* f2        = (const float*)d_in[2];
    const float* f3        = (const float*)d_in[3];
    const float* boxes     = (const float*)d_in[4];
    const int*   image_ids = (const int*)d_in[5];
    const int*   level_ids = (const int*)d_in[6];
    float*       out       = (float*)d_out;

    const int nboxes = in_sizes[5];            // image_ids: one per box
    dim3 grid((unsigned)nboxes, 4, 1);         // 4 chunks x 64 channels
    roi_align_fwd<<<grid, 64, 0, stream>>>(f0, f1, f2, f3, boxes,
                                           image_ids, level_ids, out);
}